// TemporalEncoder_17145509446146
// MI455X (gfx1250) — compile-verified
//
#include <hip/hip_runtime.h>

// Temporal spike encoder: out[t, b, d] = (t == floor(sigmoid(x[b,d]) * (T-1)))
// Store-bandwidth-bound: 419 MB of writes vs 4 MB of reads and ~1M sigmoids.
// Strategy: 1 thread per 4 consecutive d-elements; compute spike times once,
// then stream T coalesced non-temporal B128 stores (one 16B vector per plane).

typedef __attribute__((ext_vector_type(4))) float v4f;

__global__ __launch_bounds__(256) void temporal_encoder_kernel(
    const float* __restrict__ x,
    float* __restrict__ out,
    int n4,      // number of float4 groups = B*D/4
    int plane,   // B*D floats per time plane
    int T)       // timesteps
{
    const int tid = blockIdx.x * blockDim.x + threadIdx.x;
    if (tid >= n4) return;
    const int base = tid * 4;

    // Input is read exactly once across the whole kernel -> non-temporal load.
    const v4f v = __builtin_nontemporal_load((const v4f*)(x + base));

    const float scale = (float)(T - 1);
    // spike_time = floor(sigmoid(x) * (T-1)) = floor((T-1) / (1 + exp(-x)))
    // sigmoid in (0,1) so result is in [0, T-2]; floor == trunc (non-negative).
    const int s0 = (int)floorf(scale / (1.0f + __expf(-v.x)));
    const int s1 = (int)floorf(scale / (1.0f + __expf(-v.y)));
    const int s2 = (int)floorf(scale / (1.0f + __expf(-v.z)));
    const int s3 = (int)floorf(scale / (1.0f + __expf(-v.w)));

    float* o = out + base;
    #pragma unroll 4
    for (int t = 0; t < T; ++t) {
        v4f r;
        r.x = (t == s0) ? 1.0f : 0.0f;
        r.y = (t == s1) ? 1.0f : 0.0f;
        r.z = (t == s2) ? 1.0f : 0.0f;
        r.w = (t == s3) ? 1.0f : 0.0f;
        // Output (419 MB) >> L2 (192 MB): non-temporal streaming store.
        __builtin_nontemporal_store(r, (v4f*)o);
        o += plane;   // next time plane (4 MB stride; coalesced across lanes)
    }
}

extern "C" void kernel_launch(void* const* d_in, const int* in_sizes, int n_in,
                              void* d_out, int out_size, void* d_ws, size_t ws_size,
                              hipStream_t stream) {
    const float* x = (const float*)d_in[0];   // continuous_input [B, D] float32
    // d_in[1] is the 'timesteps' scalar on device; we derive T host-side
    // from out_size instead (graph-capture safe, deterministic).
    float* out = (float*)d_out;

    const int n  = in_sizes[0];       // B*D = 1,048,576
    const int T  = out_size / n;      // = 100
    const int n4 = n / 4;             // D=4096 -> divisible by 4

    const int block = 256;            // 8 wave32 per workgroup
    const int grid  = (n4 + block - 1) / block;   // 1024 blocks

    temporal_encoder_kernel<<<grid, block, 0, stream>>>(x, out, n4, n, T);
}